// GNN_73787538145803
// MI455X (gfx1250) — compile-verified
//
#include <hip/hip_runtime.h>

typedef __attribute__((ext_vector_type(2))) float v2f;
typedef __attribute__((ext_vector_type(8))) float v8f;

// ---------------------------------------------------------------------------
// Utility: zero a float buffer
// ---------------------------------------------------------------------------
__global__ void gnn_zero_f32(float* __restrict__ p, int n) {
  int i = blockIdx.x * blockDim.x + threadIdx.x;
  if (i < n) p[i] = 0.0f;
}

// ---------------------------------------------------------------------------
// Degree accumulation: deg[dst[e]] += 1
// ---------------------------------------------------------------------------
__global__ void gnn_deg_accum(const int* __restrict__ dst, float* __restrict__ deg, int E) {
  int e = blockIdx.x * blockDim.x + threadIdx.x;
  if (e < E) atomicAdd(&deg[dst[e]], 1.0f);
}

// deg += 1 (self loop); dinv = rsqrt(deg); invdeg = 1/deg
__global__ void gnn_deg_final(const float* __restrict__ deg, float* __restrict__ dinv,
                              float* __restrict__ invdeg, int N) {
  int i = blockIdx.x * blockDim.x + threadIdx.x;
  if (i < N) {
    float d = deg[i] + 1.0f;
    dinv[i]   = rsqrtf(d);
    invdeg[i] = 1.0f / d;
  }
}

// ---------------------------------------------------------------------------
// fp32 GEMM via V_WMMA_F32_16X16X4_F32; KT, NT, NOUT compile-time so every
// load/store in the unrolled body is base-pointer + immediate offset.
// One wave computes a 16 x (NT*16) output slab; the A fragment
// (one global_load_b64) feeds NT WMMAs per K-step.
// Lane layout (ISA 7.12.2): lanes 0-15 hold A[M=l][K=k0,k0+1]; lanes 16-31
// hold A[M=l][K=k0+2,k0+3]; B mirrored; C/D VGPR i -> row i + 8*(lane>=16).
// ---------------------------------------------------------------------------
template <int KT, int NT, int NOUT>
__global__ __launch_bounds__(32)
void gnn_lin_wmma_fixed(const float* __restrict__ A, const float* __restrict__ W,
                        const float* __restrict__ bias, float* __restrict__ C,
                        int relu) {
  constexpr int GROUPS = NOUT / (NT * 16);
  int tile = blockIdx.x;
  int tm   = tile / GROUPS;
  int tg   = tile - tm * GROUPS;

  int lane = threadIdx.x & 31;
  int hl   = lane >> 4;
  int l    = lane & 15;

  int col0 = tg * (NT * 16) + l;
  const float* Ap = A + (size_t)(tm * 16 + l) * KT + hl * 2;
  const float* Wp = W + (size_t)(hl * 2) * NOUT + col0;

  v8f acc[NT];
#pragma unroll
  for (int t = 0; t < NT; ++t) acc[t] = (v8f){};

#pragma unroll 4
  for (int k0 = 0; k0 < KT; k0 += 4) {
    v2f a = *(const v2f*)(Ap + k0);                 // 8B-aligned global_load_b64
    const float* wrow = Wp + k0 * NOUT;
#pragma unroll
    for (int t = 0; t < NT; ++t) {
      v2f b;
      b.x = wrow[t * 16];                           // immediate offsets
      b.y = wrow[t * 16 + NOUT];
      acc[t] = __builtin_amdgcn_wmma_f32_16x16x4_f32(
          false, a, false, b, (short)0, acc[t], false, false);
    }
  }

  float* Cp = C + (size_t)(tm * 16 + hl * 8) * NOUT + col0;
#pragma unroll
  for (int t = 0; t < NT; ++t) {
    float bv = bias ? bias[col0 + t * 16] : 0.0f;
#pragma unroll
    for (int i = 0; i < 8; ++i) {
      float v = acc[t][i] + bv;
      if (relu) v = fmaxf(v, 0.0f);
      Cp[i * NOUT + t * 16] = v;                    // immediate offsets
    }
  }
}

// K = 2 special case (layer 1: x[N x 2] @ W1[2 x NOUT]); single WMMA per
// tile, upper half-wave (K = 2,3) contributes zeros.
template <int NOUT>
__global__ __launch_bounds__(32)
void gnn_lin_wmma_k2(const float* __restrict__ A, const float* __restrict__ W,
                     float* __restrict__ C) {
  constexpr int TILES_N = NOUT / 16;
  int tile = blockIdx.x;
  int tm   = tile / TILES_N;
  int tn   = tile - tm * TILES_N;

  int lane = threadIdx.x & 31;
  int hl   = lane >> 4;
  int l    = lane & 15;
  int col  = tn * 16 + l;

  v2f a = (v2f){0.0f, 0.0f};
  v2f b = (v2f){0.0f, 0.0f};
  if (hl == 0) {
    a = *(const v2f*)(A + (size_t)(tm * 16 + l) * 2);
    b.x = W[col];
    b.y = W[NOUT + col];
  }
  v8f c = {};
  c = __builtin_amdgcn_wmma_f32_16x16x4_f32(false, a, false, b, (short)0, c,
                                            false, false);
  float* Cp = C + (size_t)(tm * 16 + hl * 8) * NOUT + col;
#pragma unroll
  for (int i = 0; i < 8; ++i)
    Cp[i * NOUT] = c[i];
}

// ---------------------------------------------------------------------------
// Edge scatter-add: agg[dst] += h[src] * (dinv[src]*dinv[dst]).
// One wave per edge; each lane handles 4 features: float4 gather
// (global_load_b128) + 4 global_atomic_add_f32. h (51 MB) is L2-resident.
// ---------------------------------------------------------------------------
template <int HH>
__global__ __launch_bounds__(32)
void gnn_edge_agg(const float* __restrict__ h, const int* __restrict__ src,
                  const int* __restrict__ dst, const float* __restrict__ dinv,
                  float* __restrict__ agg) {
  int e = blockIdx.x;
  int s = src[e], d = dst[e];
  float norm = dinv[s] * dinv[d];
  int j = (threadIdx.x & 31) * 4;
  const float4 hv = *(const float4*)(h + (size_t)s * HH + j);
  float* base = agg + (size_t)d * HH + j;
  atomicAdd(base + 0, hv.x * norm);
  atomicAdd(base + 1, hv.y * norm);
  atomicAdd(base + 2, hv.z * norm);
  atomicAdd(base + 3, hv.w * norm);
}

// out = relu(agg + h * invdeg[n] + bias[j]), float4, out may alias agg
template <int HH>
__global__ void gnn_combine(const float* __restrict__ agg, const float* __restrict__ h,
                            const float* __restrict__ invdeg, const float* __restrict__ bias,
                            float* __restrict__ out, int N) {
  int q = blockIdx.x * blockDim.x + threadIdx.x;      // float4 index
  int total = (N * HH) >> 2;
  if (q < total) {
    int idx = q * 4;
    int n = idx / HH;                                 // HH pow2 -> shift
    int j = idx & (HH - 1);
    float4 av = *(const float4*)(agg + idx);
    float4 hv = *(const float4*)(h + idx);
    float4 bv = *(const float4*)(bias + j);
    float id  = invdeg[n];
    float4 r;
    r.x = fmaxf(av.x + hv.x * id + bv.x, 0.0f);
    r.y = fmaxf(av.y + hv.y * id + bv.y, 0.0f);
    r.z = fmaxf(av.z + hv.z * id + bv.z, 0.0f);
    r.w = fmaxf(av.w + hv.w * id + bv.w, 0.0f);
    *(float4*)(out + idx) = r;
  }
}

// ---------------------------------------------------------------------------
// Mean pool: sums[batch[n]] += h[n]; cnt[batch[n]] += 1. One wave per node.
// ---------------------------------------------------------------------------
template <int HH>
__global__ __launch_bounds__(32)
void gnn_pool_accum(const float* __restrict__ h, const int* __restrict__ batch,
                    float* __restrict__ sums, float* __restrict__ cnt) {
  int n = blockIdx.x;
  int g = batch[n];
  int j = (threadIdx.x & 31) * 4;
  const float4 hv = *(const float4*)(h + (size_t)n * HH + j);
  float* base = sums + (size_t)g * HH + j;
  atomicAdd(base + 0, hv.x);
  atomicAdd(base + 1, hv.y);
  atomicAdd(base + 2, hv.z);
  atomicAdd(base + 3, hv.w);
  if (threadIdx.x == 0) atomicAdd(&cnt[g], 1.0f);
}

template <int HH>
__global__ void gnn_pool_div(float* __restrict__ sums, const float* __restrict__ cnt,
                             int G) {
  int idx = blockIdx.x * blockDim.x + threadIdx.x;
  if (idx < G * HH) {
    int g = idx / HH;
    sums[idx] = sums[idx] / fmaxf(cnt[g], 1.0f);
  }
}

// out[g] = dot(h[g,:], W[:,0]) + b[0]
template <int HH>
__global__ void gnn_fc2(const float* __restrict__ h, const float* __restrict__ W,
                        const float* __restrict__ b, float* __restrict__ out, int G) {
  int g = blockIdx.x * blockDim.x + threadIdx.x;
  if (g < G) {
    float s = b[0];
    const float* hr = h + (size_t)g * HH;
#pragma unroll 4
    for (int j = 0; j < HH; ++j) s += hr[j] * W[j];
    out[g] = s;
  }
}

// ---------------------------------------------------------------------------
// Launch (shapes fixed by the reference: H = 128, IN = 2)
// ---------------------------------------------------------------------------
extern "C" void kernel_launch(void* const* d_in, const int* in_sizes, int n_in,
                              void* d_out, int out_size, void* d_ws, size_t ws_size,
                              hipStream_t stream) {
  const float* x     = (const float*)d_in[0];
  const int*   esrc  = (const int*)d_in[1];
  const int*   edst  = (const int*)d_in[2];
  const int*   batch = (const int*)d_in[3];
  // d_in[4] = num_graphs (device scalar) — use out_size on host instead
  const float* W1    = (const float*)d_in[5];
  const float* b1    = (const float*)d_in[6];
  const float* W2    = (const float*)d_in[7];
  const float* b2    = (const float*)d_in[8];
  const float* fc1W  = (const float*)d_in[9];
  const float* fc1b  = (const float*)d_in[10];
  const float* fc2W  = (const float*)d_in[11];
  const float* fc2b  = (const float*)d_in[12];

  constexpr int H = 128;                 // == in_sizes[6]
  const int IN = in_sizes[5] / H;        // 2
  const int N  = in_sizes[0] / IN;       // 100000 (multiple of 16)
  const int E  = in_sizes[1];            // 1600000
  const int G  = out_size;               // 256

  // Workspace layout (floats)
  float* ws     = (float*)d_ws;
  float* deg    = ws;                        // N
  float* dinv   = deg    + N;                // N
  float* invdeg = dinv   + N;                // N
  float* Blin   = invdeg + N;                // N*H
  float* Bagg   = Blin   + (size_t)N * H;    // N*H
  float* sums   = Bagg   + (size_t)N * H;    // G*H
  float* cnt    = sums   + (size_t)G * H;    // G
  float* fc1o   = cnt    + G;                // G*H

  const int NH = N * H;
  const int threads = 256;
  auto blocks = [&](int n) { return (n + threads - 1) / threads; };

  constexpr int NT       = 4;
  constexpr int GROUPS_N = (H / 16) / NT;    // 2

  // --- degree / normalization ---
  gnn_zero_f32<<<blocks(N), threads, 0, stream>>>(deg, N);
  gnn_deg_accum<<<blocks(E), threads, 0, stream>>>(edst, deg, E);
  gnn_deg_final<<<blocks(N), threads, 0, stream>>>(deg, dinv, invdeg, N);

  // --- GCN layer 1 (K = IN = 2) ---
  gnn_lin_wmma_k2<H><<<(N / 16) * (H / 16), 32, 0, stream>>>(x, W1, Blin);
  gnn_zero_f32<<<blocks(NH), threads, 0, stream>>>(Bagg, NH);
  gnn_edge_agg<H><<<E, 32, 0, stream>>>(Blin, esrc, edst, dinv, Bagg);
  gnn_combine<H><<<blocks(NH / 4), threads, 0, stream>>>(Bagg, Blin, invdeg, b1, Bagg, N);

  // --- GCN layer 2 (K = H = 128) ---
  gnn_lin_wmma_fixed<H, NT, H><<<(N / 16) * GROUPS_N, 32, 0, stream>>>(
      Bagg, W2, nullptr, Blin, 0);
  gnn_zero_f32<<<blocks(NH), threads, 0, stream>>>(Bagg, NH);
  gnn_edge_agg<H><<<E, 32, 0, stream>>>(Blin, esrc, edst, dinv, Bagg);
  gnn_combine<H><<<blocks(NH / 4), threads, 0, stream>>>(Bagg, Blin, invdeg, b2, Bagg, N);

  // --- global mean pool ---
  gnn_zero_f32<<<blocks(G * H + G), threads, 0, stream>>>(sums, G * H + G);
  gnn_pool_accum<H><<<N, 32, 0, stream>>>(Bagg, batch, sums, cnt);
  gnn_pool_div<H><<<blocks(G * H), threads, 0, stream>>>(sums, cnt, G);

  // --- MLP head: fc1 (WMMA, bias + ReLU fused), fc2 dot ---
  gnn_lin_wmma_fixed<H, NT, H><<<(G / 16) * GROUPS_N, 32, 0, stream>>>(
      sums, fc1W, fc1b, fc1o, 1);
  gnn_fc2<H><<<blocks(G), threads, 0, stream>>>(fc1o, fc2W, fc2b, (float*)d_out, G);
}